// GGS_1090921693818
// MI455X (gfx1250) — compile-verified
//
#include <hip/hip_runtime.h>
#include <stdint.h>

// Problem constants (from reference)
#define B_  8
#define C_  64
#define H_  128
#define W_  128
#define HS_ 256
#define WS_ 256

// Tiling
#define TY 8
#define TX 32
#define HALOY (TY + 2)          // 10
#define HALOX (TX + 2)          // 34
#define PLANE (HALOY * HALOX)   // 340
#define CH 32                   // channels per chunk
#define NCHUNK (C_ / CH)        // 2
#define NTHREADS 256

#if defined(__has_builtin)
#  if __has_builtin(__builtin_amdgcn_tensor_load_to_lds)
#    define HAVE_TDM 1
#  endif
#  if __has_builtin(__builtin_amdgcn_wmma_f32_16x16x4_f32)
#    define HAVE_WMMA 1
#  endif
#endif
#ifndef HAVE_TDM
#  define HAVE_TDM 0
#endif
#ifndef HAVE_WMMA
#  define HAVE_WMMA 0
#endif

typedef unsigned int v4u __attribute__((ext_vector_type(4)));
typedef int          v8i __attribute__((ext_vector_type(8)));
typedef int          v4i __attribute__((ext_vector_type(4)));
typedef float        v2f __attribute__((ext_vector_type(2)));
typedef float        v8f __attribute__((ext_vector_type(8)));

// ---------------------------------------------------------------------------
// Kernel 1: channel mean.  mean[b,y,x] = (1/64) * sum_c x[b,c,y,x]
// ---------------------------------------------------------------------------
__global__ __launch_bounds__(NTHREADS)
void mean_kernel(const float* __restrict__ x, float* __restrict__ mean) {
    int p = blockIdx.x * NTHREADS + threadIdx.x;     // 0 .. B*H*W-1
    if (p >= B_ * H_ * W_) return;
    int b   = p >> 14;                                // H*W = 16384
    int rem = p & 16383;
    const float* xp = x + (((size_t)b * C_) << 14) + rem;
    float s = 0.f;
#pragma unroll 8
    for (int c = 0; c < C_; ++c) s += xp[(size_t)c << 14];
    mean[p] = s * (1.0f / 64.0f);
}

// indicator: does 5x5 tap q contribute to effective 3x3 cell (dr,dc)
// for output-parity (pi,pj) and output channel o?
__device__ __forceinline__ float indf(int pi, int pj, int o,
                                      int dr, int dc, int q, bool valid) {
    if (q >= 50) return 0.f;
    int o2 = q / 25, rr = q % 25;
    int u = rr / 5, v = rr % 5;
    // floor((pi+u-2)/2) == dr-1  <=>  ((pi+u)>>1) == dr   (pi+u in [0,5])
    return (valid && o2 == o && ((pi + u) >> 1) == dr && ((pj + v) >> 1) == dc)
               ? 1.f : 0.f;
}

// ---------------------------------------------------------------------------
// Kernel 2: fused offset-conv / kernel-weights / softmax / CARAFE gather.
// x halo tiles staged into LDS via the Tensor Data Mover (interior blocks)
// plus gfx1250 async global->LDS loads for halo columns / border blocks.
// The effective-conv precompute runs as a chain of V_WMMA_F32_16X16X4_F32.
// ---------------------------------------------------------------------------
__global__ __launch_bounds__(NTHREADS)
void carafe_kernel(const float* __restrict__ x,
                   const float* __restrict__ woff,   // [2][1][5][5]
                   const float* __restrict__ boff,   // [2]
                   const float* __restrict__ mean,   // [B][H][W]
                   float* __restrict__ out)          // [B][C][HS][WS]
{
    __shared__ float xtile[CH][HALOY][HALOX];   // 43520 B
    __shared__ float mt[PLANE];                 // mean halo tile
    __shared__ float weff[2][2][2][9];          // effective 3x3 conv weights

    const int tid = threadIdx.x;
    const int x0  = blockIdx.x * TX;
    const int y0  = blockIdx.y * TY;
    const int b   = blockIdx.z;

    // ---- mean halo tile (zero outside image == reference zero-padding) ----
    for (int t = tid; t < PLANE; t += NTHREADS) {
        int r  = t / HALOX, col = t - r * HALOX;
        int gy = y0 - 1 + r, gx = x0 - 1 + col;
        float v = 0.f;
        if (gy >= 0 && gy < H_ && gx >= 0 && gx < W_)
            v = mean[(b << 14) + (gy << 7) + gx];
        mt[t] = v;
    }

    // ---- weff[e] = sum_q IND[e,q] * woff[q]  (72x50 indicator matvec) ----
#if HAVE_WMMA
    if (tid < 32) {                 // wave 0, EXEC all ones (WMMA requirement)
        const int lane = tid;
        const int half = lane >> 4;             // K half: 0 -> K{0,1}, 1 -> K{2,3}
        const int n    = lane & 15;             // column / row index
        float wq[52];
#pragma unroll
        for (int q = 0; q < 50; ++q) wq[q] = woff[q];
        wq[50] = 0.f; wq[51] = 0.f;             // K padding (50 -> 52)

        for (int grp = 0; grp < 5; ++grp) {     // 5 column groups of 16 e's
            int e = grp * 16 + n;
            bool valid = e < 72;
            int e2 = valid ? e : 0;
            int dc = e2 % 3, t0 = e2 / 3;
            int dr = t0 % 3;  t0 /= 3;
            int o  = t0 % 2;  t0 /= 2;
            int pj = t0 % 2;
            int pi = t0 / 2;

            v8f acc = {0.f, 0.f, 0.f, 0.f, 0.f, 0.f, 0.f, 0.f};
#pragma unroll
            for (int kk = 0; kk < 13; ++kk) {   // K = 52 in steps of 4
                // A[m,k] = wq[k] broadcast over rows; lane half selects K pair
                v2f a;
                a.x = half ? wq[4 * kk + 2] : wq[4 * kk + 0];
                a.y = half ? wq[4 * kk + 3] : wq[4 * kk + 1];
                // B[k,n] = IND[e_n, k]
                v2f bm;
                bm.x = half ? indf(pi, pj, o, dr, dc, 4 * kk + 2, valid)
                            : indf(pi, pj, o, dr, dc, 4 * kk + 0, valid);
                bm.y = half ? indf(pi, pj, o, dr, dc, 4 * kk + 3, valid)
                            : indf(pi, pj, o, dr, dc, 4 * kk + 1, valid);
                acc = __builtin_amdgcn_wmma_f32_16x16x4_f32(
                          false, a, false, bm, (short)0, acc, false, false);
            }
            // all D rows identical (A rows identical); row 0 lives in
            // acc[0] of lanes 0..15
            if (lane < 16 && valid)
                ((float*)weff)[e] = acc[0];
        }
    }
#else
    if (tid < 72) {
        int e  = tid;
        int dc = e % 3; e /= 3;
        int dr = e % 3; e /= 3;
        int o  = e % 2; e /= 2;
        int pj = e % 2; e /= 2;
        int pi = e;
        float s = 0.f;
        for (int u = 0; u < 5; ++u) {
            if ((((pi + u) >> 1) - 1) != dr - 1) continue;
            for (int v = 0; v < 5; ++v)
                if ((((pj + v) >> 1) - 1) == dc - 1)
                    s += woff[(o * 5 + u) * 5 + v];
        }
        weff[pi][pj][o][dr * 3 + dc] = s;
    }
#endif
    __syncthreads();

    // ---- per-thread low-res pixel: softmaxed 2x2x9 weights ----
    const int lx = tid & (TX - 1);
    const int ly = tid >> 5;

    float mwin[9];
#pragma unroll
    for (int k = 0; k < 9; ++k)
        mwin[k] = mt[(ly + k / 3) * HALOX + (lx + k % 3)];
    const float mc = mwin[4];

    float g[9];
#pragma unroll
    for (int k = 0; k < 9; ++k) {
        float d = mwin[k] - mc;
        g[k] = 1.0f / (d * d + 1.0f);
    }

    const float b0 = boff[0], b1 = boff[1];
    float wk[4][9];
#pragma unroll
    for (int pi = 0; pi < 2; ++pi) {
#pragma unroll
        for (int pj = 0; pj < 2; ++pj) {
            float o0 = b0, o1 = b1;
#pragma unroll
            for (int k = 0; k < 9; ++k) {
                o0 = fmaf(weff[pi][pj][0][k], mwin[k], o0);
                o1 = fmaf(weff[pi][pj][1][k], mwin[k], o1);
            }
            float s0 = (pi ? 0.25f : -0.25f) + 0.25f * tanhf(o0);
            float s1 = (pj ? 0.25f : -0.25f) + 0.25f * tanhf(o1);
            float sc[9], mx = -1e30f;
#pragma unroll
            for (int k = 0; k < 9; ++k) {
                float d0 = s0 - (float)(k / 3 - 1);
                float d1 = s1 - (float)(k % 3 - 1);
                float kern = 1.0f / (d0 * d0 + d1 * d1 + 0.5f);
                sc[k] = g[k] * kern;
                mx = fmaxf(mx, sc[k]);
            }
            float sum = 0.f;
#pragma unroll
            for (int k = 0; k < 9; ++k) { float e = __expf(sc[k] - mx); sc[k] = e; sum += e; }
            float inv = 1.0f / sum;
#pragma unroll
            for (int k = 0; k < 9; ++k) wk[pi * 2 + pj][k] = sc[k] * inv;
        }
    }

    const float* xb   = x + (((size_t)(b * C_)) << 14);
    float2*      out2 = (float2*)out;
    const int    oy   = 2 * (y0 + ly);
    const int    ox   = x0 + lx;
    const bool   interior_y = (y0 > 0) && (y0 + TY < H_);   // halo rows in-bounds

    for (int chunk = 0; chunk < NCHUNK; ++chunk) {
        __syncthreads();                              // WAR: prior readers done

#if HAVE_TDM
        if (interior_y) {
            // ---- TDM: one 3D tile load (32 cols x 10 rows x 32 channels).
            // LDS pad: after every 32 data dwords skip 2 dwords -> output
            // lands exactly at xtile[c][r][1..32], skipping halo cols 33/0.
            if (tid < 32) {
                uint64_t ga = (uint64_t)(uintptr_t)
                    (xb + (((size_t)(chunk * CH)) << 14) + ((y0 - 1) << 7) + x0);
                unsigned lds0 = (unsigned)(uintptr_t)&xtile[0][0][1];
                v4u g0 = { 1u,                                // count=1
                           lds0,                              // lds_addr
                           (unsigned)ga,                      // global_addr[31:0]
                           (unsigned)((ga >> 32) & 0x1FFFFFFu) | (2u << 30) }; // [56:32]|type=2
                v8i g1 = { (int)0x03120000u,   // data_size=4B, pad_en, intv=32dw, amt=2dw
                           (int)0xFFFF0000u,   // tensor_dim0 lo16 (huge: no OOB)
                           (int)0xFFFF7FFFu,   // tensor_dim1 lo16 | tensor_dim0 hi16
                           (int)((32 << 16) | 0x7FFF),  // tile_dim0=32 | td1 hi16
                           (int)((CH << 16) | HALOY),   // tile_dim2=32 | tile_dim1=10
                           (int)W_,                     // dim0 stride = 128
                           (int)0x40000000,             // dim1 stride lo16(16384)<<16
                           0 };
                v4i g2 = { 0x7FFFFFFF, 0, 0, 0 };       // tensor_dim2 huge
                v4i g3 = { 0, 0, 0, 0 };
                v8i g4 = { 0, 0, 0, 0, 0, 0, 0, 0 };    // terminating NULL descriptor
                __builtin_amdgcn_tensor_load_to_lds(g0, g1, g2, g3, g4, 0);
            }
            // ---- halo columns (LDS col 0 and 33) via async loads ----
            for (int t = tid; t < CH * HALOY * 2; t += NTHREADS) {
                int side = t & 1;
                int idx  = t >> 1;
                int r    = idx % HALOY;
                int c    = idx / HALOY;
                int col  = side ? (HALOX - 1) : 0;
                int gx   = side ? (x0 + TX) : (x0 - 1);
                int gy   = y0 - 1 + r;                       // in-bounds here
                unsigned lds = (unsigned)(uintptr_t)&xtile[c][r][col];
                if (gx >= 0 && gx < W_) {
                    uint64_t ga = (uint64_t)(uintptr_t)
                        (xb + (((size_t)(chunk * CH + c)) << 14) + (gy << 7) + gx);
                    asm volatile("global_load_async_to_lds_b32 %0, %1, off"
                                 :: "v"(lds), "v"(ga) : "memory");
                } else {
                    xtile[c][r][col] = 0.f;
                }
            }
            asm volatile("s_wait_asynccnt 0" ::: "memory");
            if (tid < 32) __builtin_amdgcn_s_wait_tensorcnt(0);
        } else
#endif
        {
            // ---- border blocks: per-element async loads + zero padding ----
            for (int t = tid; t < CH * PLANE; t += NTHREADS) {
                int c   = t / PLANE;
                int rem = t - c * PLANE;
                int r   = rem / HALOX, col = rem - r * HALOX;
                int gy  = y0 - 1 + r,  gx  = x0 - 1 + col;
                unsigned lds = (unsigned)(uintptr_t)&(((float*)xtile)[t]);
                if (gy >= 0 && gy < H_ && gx >= 0 && gx < W_) {
                    uint64_t ga = (uint64_t)(uintptr_t)
                        (xb + (((size_t)(chunk * CH + c)) << 14) + (gy << 7) + gx);
                    asm volatile("global_load_async_to_lds_b32 %0, %1, off"
                                 :: "v"(lds), "v"(ga) : "memory");
                } else {
                    ((float*)xtile)[t] = 0.f;
                }
            }
            asm volatile("s_wait_asynccnt 0" ::: "memory");
        }
        __syncthreads();

        // ---- 9-tap weighted gather, 32 channels, coalesced b64 stores ----
#pragma unroll 4
        for (int c = 0; c < CH; ++c) {
            float p[9];
#pragma unroll
            for (int k = 0; k < 9; ++k)
                p[k] = xtile[c][ly + k / 3][lx + k % 3];
            float o00 = 0.f, o01 = 0.f, o10 = 0.f, o11 = 0.f;
#pragma unroll
            for (int k = 0; k < 9; ++k) {
                o00 = fmaf(wk[0][k], p[k], o00);
                o01 = fmaf(wk[1][k], p[k], o01);
                o10 = fmaf(wk[2][k], p[k], o10);
                o11 = fmaf(wk[3][k], p[k], o11);
            }
            int gc = chunk * CH + c;
            size_t rb = (((size_t)(b * C_ + gc) * HS_) + oy) * (WS_ / 2) + ox;
            out2[rb]           = make_float2(o00, o01);   // row 2y
            out2[rb + WS_ / 2] = make_float2(o10, o11);   // row 2y+1
        }
    }
}

// ---------------------------------------------------------------------------
extern "C" void kernel_launch(void* const* d_in, const int* in_sizes, int n_in,
                              void* d_out, int out_size, void* d_ws, size_t ws_size,
                              hipStream_t stream) {
    const float* x  = (const float*)d_in[0];
    const float* wo = (const float*)d_in[1];
    const float* bo = (const float*)d_in[2];
    float* out  = (float*)d_out;
    float* mean = (float*)d_ws;                       // B*H*W floats = 512 KB

    mean_kernel<<<(B_ * H_ * W_) / NTHREADS, NTHREADS, 0, stream>>>(x, mean);

    dim3 grid(W_ / TX, H_ / TY, B_);                  // 4 x 16 x 8 = 512 blocks
    carafe_kernel<<<grid, NTHREADS, 0, stream>>>(x, wo, bo, mean, out);
}